// TemporalGNN_13743895347604
// MI455X (gfx1250) — compile-verified
//
#include <hip/hip_runtime.h>
#include <hip/hip_bf16.h>
#include <math.h>

typedef __attribute__((ext_vector_type(2))) float v2f;
typedef __attribute__((ext_vector_type(8))) float v8f;

#define GF   8    // features
#define GH   32   // hidden
#define GP   12   // time steps
#define XROW 96   // F*P floats per node

// ---------------- graph aggregation kernels ----------------

__global__ void k_init_deg(float* deg, int N) {
    int i = blockIdx.x * blockDim.x + threadIdx.x;
    if (i < N) deg[i] = 1.0f;           // self loop
}

__global__ void k_count_deg(const int* ei, float* deg, int E) {
    int e = blockIdx.x * blockDim.x + threadIdx.x;
    if (e < E) atomicAdd(&deg[ei[E + e]], 1.0f);
}

__global__ void k_rsqrt(float* deg, int N) {
    int i = blockIdx.x * blockDim.x + threadIdx.x;
    if (i < N) deg[i] = rsqrtf(deg[i]);
}

// aggX[n, :, :] = x[n, :, :] * dinv[n]^2   (self-loop term, also zero-initializes)
__global__ void k_self_agg(const float* __restrict__ x, const float* __restrict__ dinv,
                           float* __restrict__ aggX, int N) {
    int i = blockIdx.x * blockDim.x + threadIdx.x;
    if (i < N * XROW) {
        int n = i / XROW;
        float dv = dinv[n];
        aggX[i] = x[i] * dv * dv;
    }
}

// one thread per (edge, feature): 12 contiguous time values, scatter-add into dst
__global__ void k_edge_agg(const float* __restrict__ x, const int* __restrict__ ei,
                           const float* __restrict__ dinv, float* __restrict__ aggX, int E) {
    int t = blockIdx.x * blockDim.x + threadIdx.x;
    if (t >= E * GF) return;
    int e = t >> 3, f = t & 7;
    int s = ei[e], d = ei[E + e];
    float nrm = dinv[s] * dinv[d];
    const float* xs = x + (size_t)s * XROW + f * GP;
    float* ad = aggX + (size_t)d * XROW + f * GP;
#pragma unroll
    for (int p = 0; p < GP; ++p)
        atomicAdd(&ad[p], xs[p] * nrm);
}

// ---------------- fused GRU recurrence + output head ----------------
// 128 threads = 4 wave32; each wave owns 16 nodes for all 12 steps.
// Lane l: node m = l&15, owns H/acc columns [16*half, 16*half+16), half = l>>4.

#define AS 68   // A-tile row stride (16 rows x 64 cols, padded: banks row*68%64 distinct)
#define DS 36   // D-tile row stride (16 rows x 32 cols, padded)

__global__ __launch_bounds__(128) void k_gru(
    const float* __restrict__ aggX, const float* __restrict__ att,
    const float* __restrict__ Wz, const float* __restrict__ bz,
    const float* __restrict__ Wr, const float* __restrict__ br,
    const float* __restrict__ Wh, const float* __restrict__ bh,
    const float* __restrict__ Lz, const float* __restrict__ lbz,
    const float* __restrict__ Lr, const float* __restrict__ lbr,
    const float* __restrict__ Lh, const float* __restrict__ lbh,
    const float* __restrict__ Wout, const float* __restrict__ bout,
    float* __restrict__ out, int N)
{
    // L matrices stored ONLY in pair-swizzled form:
    //   sw[q*64 + 2n + 0] = L[2q+0][n],  sw[q*64 + 2n + 1] = L[2q+1][n]
    // so a lane's WMMA B-fragment (rows kb,kb+1 at column n) is one aligned b64.
    __shared__ __align__(16) float sLz[64 * 32], sLr[64 * 32], sLh[64 * 32];
    __shared__ float sWz[GF * 32], sWr[GF * 32], sWh[GF * 32];
    __shared__ float sbz[32], sbr[32], sbh[32], slbz[32], slbr[32], slbh[32];
    __shared__ float sWout[32 * GF], sbout[GF], sAtt[GP];
    __shared__ __align__(16) float sA[4][16 * AS];
    __shared__ float sDt[4][16 * DS];

    int tid = threadIdx.x;
    for (int i = tid; i < 64 * 32; i += 128) {
        int q = i >> 6, t = i & 63, n = t >> 1, r = 2 * q + (t & 1);
        int src = r * 32 + n;
        sLz[i] = Lz[src]; sLr[i] = Lr[src]; sLh[i] = Lh[src];
    }
    for (int i = tid; i < GF * 32; i += 128) { sWz[i] = Wz[i]; sWr[i] = Wr[i]; sWh[i] = Wh[i]; }
    if (tid < 32) {
        sbz[tid] = bz[tid]; sbr[tid] = br[tid]; sbh[tid] = bh[tid];
        slbz[tid] = lbz[tid]; slbr[tid] = lbr[tid]; slbh[tid] = lbh[tid];
    }
    for (int i = tid; i < 32 * GF; i += 128) sWout[i] = Wout[i];
    if (tid < GF) sbout[tid] = bout[tid];
    if (tid < GP) sAtt[tid] = att[tid];
    __syncthreads();

    const int wave = tid >> 5;
    const int lane = tid & 31;
    const int m = lane & 15;
    const int half = lane >> 4;
    const int node = blockIdx.x * 64 + wave * 16 + m;
    const int nodeC = node < N ? node : N - 1;

    // softmax(attention) pieces (tiny, redundant per lane)
    float amax = sAtt[0];
    for (int p = 1; p < GP; ++p) amax = fmaxf(amax, sAtt[p]);
    float asum = 0.f;
    for (int p = 0; p < GP; ++p) asum += __expf(sAtt[p] - amax);
    const float ainv = 1.0f / asum;

    float H[16], acc[16];
#pragma unroll
    for (int j = 0; j < 16; ++j) { H[j] = 0.f; acc[j] = 0.f; }

    float* A = sA[wave];
    float* D = sDt[wave];

    // 16x64 @ 64x32 via chained V_WMMA_F32_16X16X4_F32.
    // All fragments are bulk-loaded first (single dscnt wait), then two
    // independent accumulation chains (N-tiles 0/1) are interleaved so the
    // matrix pipe never stalls on the serial C dependency.
    auto matmul = [&](const float* __restrict__ Bw, float* outv) {
        v2f a[16], b0[16], b1[16];
#pragma unroll
        for (int k = 0; k < 16; ++k) {
            const int kb = 4 * k + 2 * half;       // ISA A-layout: lanes16-31 hold K+2
            a[k]  = *(const v2f*)&A[m * AS + kb];  // rows: lane = M, pair (kb,kb+1)
            const int q = 2 * k + half;            // swizzled pair row
            b0[k] = *(const v2f*)&Bw[q * 64 + 2 * m];
            b1[k] = *(const v2f*)&Bw[q * 64 + 2 * (16 + m)];
        }
        v8f c0 = {}, c1 = {};
#pragma unroll
        for (int k = 0; k < 16; ++k) {
            c0 = __builtin_amdgcn_wmma_f32_16x16x4_f32(
                     false, a[k], false, b0[k], (short)0, c0, false, false);
            c1 = __builtin_amdgcn_wmma_f32_16x16x4_f32(
                     false, a[k], false, b1[k], (short)0, c1, false, false);
        }
#pragma unroll
        for (int r = 0; r < 8; ++r) {              // D: VGPR r = row r (lanes 0-15) / r+8
            D[(r + 8 * half) * DS + m]      = c0[r];
            D[(r + 8 * half) * DS + 16 + m] = c1[r];
        }
        asm volatile("s_wait_dscnt 0" ::: "memory");
#pragma unroll
        for (int i = 0; i < 16; ++i)
            outv[i] = D[m * DS + 16 * half + i];
    };

#pragma unroll 1
    for (int p = 0; p < GP; ++p) {
        const float prob = __expf(sAtt[p] - amax) * ainv;
        float xa[GF];
#pragma unroll
        for (int f = 0; f < GF; ++f)
            xa[f] = aggX[(size_t)nodeC * XROW + f * GP + p];

        // ---- gate Z: A = [gcn_z | H] ----
#pragma unroll
        for (int i = 0; i < 16; ++i) {
            const int col = 16 * half + i;
            float g = sbz[col];
#pragma unroll
            for (int f = 0; f < GF; ++f) g += xa[f] * sWz[f * 32 + col];
            A[m * AS + col] = g;
            A[m * AS + 32 + col] = H[i];
        }
        asm volatile("s_wait_dscnt 0" ::: "memory");
        float Zv[16];
        matmul(sLz, Zv);

        // ---- gate R: reuse H half, rewrite gcn half ----
#pragma unroll
        for (int i = 0; i < 16; ++i) {
            const int col = 16 * half + i;
            float g = sbr[col];
#pragma unroll
            for (int f = 0; f < GF; ++f) g += xa[f] * sWr[f * 32 + col];
            A[m * AS + col] = g;
        }
        asm volatile("s_wait_dscnt 0" ::: "memory");
        float Rv[16];
        matmul(sLr, Rv);

#pragma unroll
        for (int i = 0; i < 16; ++i) {
            const int col = 16 * half + i;
            Zv[i] = 1.0f / (1.0f + __expf(-(Zv[i] + slbz[col])));
            Rv[i] = 1.0f / (1.0f + __expf(-(Rv[i] + slbr[col])));
        }

        // ---- candidate: A = [gcn_h | R*H] ----
#pragma unroll
        for (int i = 0; i < 16; ++i) {
            const int col = 16 * half + i;
            float g = sbh[col];
#pragma unroll
            for (int f = 0; f < GF; ++f) g += xa[f] * sWh[f * 32 + col];
            A[m * AS + col] = g;
            A[m * AS + 32 + col] = Rv[i] * H[i];
        }
        asm volatile("s_wait_dscnt 0" ::: "memory");
        float Hc[16];
        matmul(sLh, Hc);

#pragma unroll
        for (int i = 0; i < 16; ++i) {
            const int col = 16 * half + i;
            const float ht = tanhf(Hc[i] + slbh[col]);
            H[i] = Zv[i] * H[i] + (1.0f - Zv[i]) * ht;
            acc[i] += prob * H[i];
        }
    }

    // ---- output head: relu(acc) @ Wout + bout ----
#pragma unroll
    for (int i = 0; i < 16; ++i)
        D[m * DS + 16 * half + i] = fmaxf(acc[i], 0.f);
    asm volatile("s_wait_dscnt 0" ::: "memory");

    if (half == 0 && node < N) {
        for (int o = 0; o < GF; ++o) {
            float s = sbout[o];
#pragma unroll
            for (int j = 0; j < 32; ++j) s += D[m * DS + j] * sWout[j * GF + o];
            out[(size_t)node * GF + o] = s;
        }
    }
}

// ---------------- launcher ----------------

extern "C" void kernel_launch(void* const* d_in, const int* in_sizes, int n_in,
                              void* d_out, int out_size, void* d_ws, size_t ws_size,
                              hipStream_t stream) {
    const float* x    = (const float*)d_in[0];
    const int*   ei   = (const int*)  d_in[1];
    const float* att  = (const float*)d_in[2];
    const float* Wz   = (const float*)d_in[3];
    const float* bz   = (const float*)d_in[4];
    const float* Wr   = (const float*)d_in[5];
    const float* br   = (const float*)d_in[6];
    const float* Wh   = (const float*)d_in[7];
    const float* bh   = (const float*)d_in[8];
    const float* Lz   = (const float*)d_in[9];
    const float* lbz  = (const float*)d_in[10];
    const float* Lr   = (const float*)d_in[11];
    const float* lbr  = (const float*)d_in[12];
    const float* Lh   = (const float*)d_in[13];
    const float* lbh  = (const float*)d_in[14];
    const float* Wout = (const float*)d_in[15];
    const float* bout = (const float*)d_in[16];
    float* out = (float*)d_out;

    const int N = in_sizes[0] / XROW;
    const int E = in_sizes[1] / 2;

    float* deg  = (float*)d_ws;
    float* aggX = deg + ((N + 255) / 256) * 256;   // aligned; N*96 floats

    k_init_deg<<<(N + 255) / 256, 256, 0, stream>>>(deg, N);
    k_count_deg<<<(E + 255) / 256, 256, 0, stream>>>(ei, deg, E);
    k_rsqrt<<<(N + 255) / 256, 256, 0, stream>>>(deg, N);
    k_self_agg<<<(N * XROW + 255) / 256, 256, 0, stream>>>(x, deg, aggX, N);
    k_edge_agg<<<((E * GF) + 255) / 256, 256, 0, stream>>>(x, ei, deg, aggX, E);
    k_gru<<<(N + 63) / 64, 128, 0, stream>>>(aggX, att, Wz, bz, Wr, br, Wh, bh,
                                             Lz, lbz, Lr, lbr, Lh, lbh,
                                             Wout, bout, out, N);
}